// PAC_72121090835151
// MI455X (gfx1250) — compile-verified
//
#include <hip/hip_runtime.h>
#include <hip/hip_bf16.h>
#include <math.h>

// ---------------- problem constants ----------------
#define BCN   16      // B*C rows
#define TLEN  2048    // samples
#define TT    1536    // cropped samples (edge=256)
#define EDGE  256
#define KB    640     // spectrum bins actually needed (covers max band edge 158.625Hz / 0.25Hz = 634.5)
#define NBAND 80      // 50 phase + 30 amp
#define PHAN  50
#define AMPN  30
#define NBINS 18

#define PI_F 3.14159265358979323846f
#define TWO_PI_F 6.28318530717958647692f

typedef float v2f __attribute__((ext_vector_type(2)));
typedef float v8f __attribute__((ext_vector_type(8)));

// ---------------- K1: narrow-band direct DFT ----------------
// Xf[bc][k] = sum_t x[bc][t] * e^{-2*pi*i*k*t/T},  k < 640.
// 10240 independent bins; argument reduced mod 2048 for sincos precision.
__global__ void __launch_bounds__(256) pac_dft_kernel(const float* __restrict__ x,
                                                      float* __restrict__ Xre,
                                                      float* __restrict__ Xim) {
    int gid = blockIdx.x * blockDim.x + threadIdx.x;
    if (gid >= BCN * KB) return;
    int bc = gid / KB;
    int k  = gid - bc * KB;
    const float* xr = x + (size_t)bc * TLEN;
    float sr = 0.0f, si = 0.0f;
    for (int t = 0; t < TLEN; ++t) {
        int m = (k * t) & (TLEN - 1);
        float ang = -TWO_PI_F * (float)m * (1.0f / (float)TLEN);
        float s, c;
        __sincosf(ang, &s, &c);
        float v = xr[t];
        sr = fmaf(v, c, sr);
        si = fmaf(v, s, si);
    }
    Xre[gid] = sr;
    Xim[gid] = si;
}

// ---------------- K2: inverse-DFT twiddle matrix ----------------
// B[k][tt] = e^{+2*pi*i*k*(tt+EDGE)/T}; shared by all 16 rows, L2-resident (7.8MB).
__global__ void __launch_bounds__(256) pac_twiddle_kernel(float* __restrict__ Bre,
                                                          float* __restrict__ Bim) {
    int gid = blockIdx.x * blockDim.x + threadIdx.x;
    if (gid >= KB * TT) return;
    int k  = gid / TT;
    int tt = gid - k * TT;
    int m = (k * (tt + EDGE)) & (TLEN - 1);
    float ang = TWO_PI_F * (float)m * (1.0f / (float)TLEN);
    float s, c;
    __sincosf(ang, &s, &c);
    Bre[gid] = c;
    Bim[gid] = s;
}

// ---------------- K3: band-masked coefficient matrices ----------------
// A[bc][band][k] = in_band(band,k) ? (2/T) * Xf[bc][k] : 0
// Band edges computed in double then rounded to f32 to match numpy linspace->astype(float32).
__global__ void __launch_bounds__(256) pac_bandcoef_kernel(const float* __restrict__ Xre,
                                                           const float* __restrict__ Xim,
                                                           float* __restrict__ Are,
                                                           float* __restrict__ Aim) {
    int gid = blockIdx.x * blockDim.x + threadIdx.x;
    if (gid >= BCN * NBAND * KB) return;
    int bc   = gid / (NBAND * KB);
    int rem  = gid - bc * (NBAND * KB);
    int band = rem / KB;
    int k    = rem - band * KB;

    double lo, hi;
    if (band < PHAN) {
        double mid = 2.0 + 18.0 * (double)band / 49.0;         // linspace(2,20,50)
        lo = 0.75 * mid;  hi = 1.25 * mid;
    } else {
        int j = band - PHAN;
        double mid = 60.0 + 81.0 * (double)j / 29.0;           // linspace(60,141,30)
        lo = 0.875 * mid; hi = 1.125 * mid;
    }
    float lof = (float)lo, hif = (float)hi;
    float f = 0.25f * (float)k;                                // FS/T = 0.25 Hz per bin
    bool inband = (f >= lof) && (f <= hif);
    float scale = inband ? (2.0f / (float)TLEN) : 0.0f;        // Hilbert x2, IFFT 1/T
    Are[gid] = scale * Xre[(size_t)bc * KB + k];
    Aim[gid] = scale * Xim[(size_t)bc * KB + k];
}

// ---------------- K4: WMMA f32 complex GEMM band synthesis ----------------
// analytic = A (80x640 cplx) * B (640x1536 cplx) per bc, tiled 16x16, K-step 4
// via V_WMMA_F32_16X16X4_F32.  Then angle -> phase-bin index, or magnitude.
__global__ void __launch_bounds__(32) pac_synth_wmma_kernel(const float* __restrict__ Are,
                                                            const float* __restrict__ Aim,
                                                            const float* __restrict__ Bre,
                                                            const float* __restrict__ Bim,
                                                            int*   __restrict__ IDX,
                                                            float* __restrict__ AMP) {
    const int lane = threadIdx.x;        // one wave per block -> EXEC all ones
    const int t0   = blockIdx.x * 16;    // 96 tiles over TT
    const int m0   = blockIdx.y * 16;    // 5 tiles over 80 bands
    const int bc   = blockIdx.z;         // 16 rows
    const int half = lane >> 4;          // 0 or 1
    const int l15  = lane & 15;

    const float* A_re = Are + (size_t)bc * NBAND * KB;
    const float* A_im = Aim + (size_t)bc * NBAND * KB;

    v8f cre = {};
    v8f cim = {};

    const int arow = m0 + l15;           // A frag: lanes 0-15 M=lane K={k0,k0+1}; 16-31 M=lane-16 K={k0+2,k0+3}
    for (int k0 = 0; k0 < KB; k0 += 4) {
        const int ka = k0 + half * 2;
        v2f a_re, a_im, a_imn, b_re, b_im;
        const float* ap  = A_re + (size_t)arow * KB + ka;
        const float* aip = A_im + (size_t)arow * KB + ka;
        a_re.x = ap[0];   a_re.y = ap[1];
        a_im.x = aip[0];  a_im.y = aip[1];
        a_imn.x = -a_im.x; a_imn.y = -a_im.y;
        const float* bp  = Bre + (size_t)ka * TT + t0 + l15;   // B frag mirrors A: row-striped
        const float* bip = Bim + (size_t)ka * TT + t0 + l15;
        b_re.x = bp[0];   b_re.y = bp[TT];
        b_im.x = bip[0];  b_im.y = bip[TT];

        // c_re += Ar*Br + (-Ai)*Bi ; c_im += Ar*Bi + Ai*Br
        cre = __builtin_amdgcn_wmma_f32_16x16x4_f32(false, a_re,  false, b_re, (short)0, cre, false, false);
        cre = __builtin_amdgcn_wmma_f32_16x16x4_f32(false, a_imn, false, b_im, (short)0, cre, false, false);
        cim = __builtin_amdgcn_wmma_f32_16x16x4_f32(false, a_re,  false, b_im, (short)0, cim, false, false);
        cim = __builtin_amdgcn_wmma_f32_16x16x4_f32(false, a_im,  false, b_re, (short)0, cim, false, false);
    }

    // C/D layout: VGPR j -> M = m0 + j + 8*half, N = t0 + (lane&15)
    const int tcol = t0 + l15;
    const float inv_width = (float)NBINS / TWO_PI_F;   // 1 / (2pi/18)
    for (int j = 0; j < 8; ++j) {
        int band = m0 + j + half * 8;
        float re = cre[j], im = cim[j];
        if (band < PHAN) {
            float ph = atan2f(im, re);
            int idx = (int)floorf((ph + PI_F) * inv_width);
            idx = idx < 0 ? 0 : (idx > NBINS - 1 ? NBINS - 1 : idx);
            IDX[((size_t)bc * PHAN + band) * TT + tcol] = idx;
        } else {
            AMP[((size_t)bc * AMPN + (band - PHAN)) * TT + tcol] = sqrtf(fmaf(re, re, im * im));
        }
    }
}

// ---------------- K5: histogram + modulation index ----------------
// One block per (bc, phase-band): LDS scatter of 30 amp rows into 18 bins, then MI.
__global__ void __launch_bounds__(256) pac_mi_kernel(const int* __restrict__ IDX,
                                                     const float* __restrict__ AMP,
                                                     float* __restrict__ out) {
    const int bc = blockIdx.x / PHAN;
    const int p  = blockIdx.x - bc * PHAN;

    __shared__ float s_cnt[NBINS];
    __shared__ float s_sum[AMPN * NBINS];
    for (int i = threadIdx.x; i < NBINS; i += 256) s_cnt[i] = 0.0f;
    for (int i = threadIdx.x; i < AMPN * NBINS; i += 256) s_sum[i] = 0.0f;
    __syncthreads();

    const int*   idxrow = IDX + ((size_t)bc * PHAN + p) * TT;
    const float* amp    = AMP + (size_t)bc * AMPN * TT;
    for (int t = threadIdx.x; t < TT; t += 256) {
        int k = idxrow[t];
        atomicAdd(&s_cnt[k], 1.0f);
        #pragma unroll 6
        for (int a = 0; a < AMPN; ++a)
            atomicAdd(&s_sum[a * NBINS + k], amp[(size_t)a * TT + t]);
    }
    __syncthreads();

    if (threadIdx.x < AMPN) {
        const int a = threadIdx.x;
        float mean[NBINS];
        float tot = 0.0f;
        for (int k = 0; k < NBINS; ++k) {
            float m = s_sum[a * NBINS + k] / fmaxf(s_cnt[k], 1e-9f);
            mean[k] = m;
            tot += m;
        }
        tot = fmaxf(tot, 1e-9f);
        float acc = 0.0f;
        for (int k = 0; k < NBINS; ++k) {
            float pr = mean[k] / tot;
            acc += pr * logf(pr + 1e-9f);
        }
        const float lognb = logf((float)NBINS);
        out[((size_t)bc * PHAN + p) * AMPN + a] = (lognb + acc) / lognb;
    }
}

// ---------------- launcher ----------------
extern "C" void kernel_launch(void* const* d_in, const int* in_sizes, int n_in,
                              void* d_out, int out_size, void* d_ws, size_t ws_size,
                              hipStream_t stream) {
    const float* x = (const float*)d_in[0];   // (2,8,1,2048) f32, row-major -> 16 rows of 2048
    float* out = (float*)d_out;               // (2,8,50,30) f32

    // workspace layout (floats); total ~22.4 MB
    float* ws  = (float*)d_ws;
    float* Xre = ws;                                  // 16*640
    float* Xim = Xre + BCN * KB;                      // 16*640
    float* Bre = Xim + BCN * KB;                      // 640*1536
    float* Bim = Bre + KB * TT;                       // 640*1536
    float* Are = Bim + KB * TT;                       // 16*80*640
    float* Aim = Are + BCN * NBAND * KB;              // 16*80*640
    float* AMP = Aim + BCN * NBAND * KB;              // 16*30*1536
    int*   IDX = (int*)(AMP + BCN * AMPN * TT);       // 16*50*1536

    (void)in_sizes; (void)n_in; (void)out_size; (void)ws_size;

    // K1: narrow-band DFT (16*640 bins)
    {
        int n = BCN * KB;
        pac_dft_kernel<<<(n + 255) / 256, 256, 0, stream>>>(x, Xre, Xim);
    }
    // K2: twiddle matrix (640 x 1536 complex)
    {
        int n = KB * TT;
        pac_twiddle_kernel<<<(n + 255) / 256, 256, 0, stream>>>(Bre, Bim);
    }
    // K3: band-masked coefficient matrices (16 x 80 x 640)
    {
        int n = BCN * NBAND * KB;
        pac_bandcoef_kernel<<<(n + 255) / 256, 256, 0, stream>>>(Xre, Xim, Are, Aim);
    }
    // K4: WMMA complex GEMM synthesis -> phase-bin indices + amplitudes
    {
        dim3 grid(TT / 16, NBAND / 16, BCN);   // 96 x 5 x 16 waves
        pac_synth_wmma_kernel<<<grid, 32, 0, stream>>>(Are, Aim, Bre, Bim, IDX, AMP);
    }
    // K5: histogram + MI (one block per (bc, phase band))
    {
        pac_mi_kernel<<<BCN * PHAN, 256, 0, stream>>>(IDX, AMP, out);
    }
}